// KNN_15925738734005
// MI455X (gfx1250) — compile-verified
//
#include <hip/hip_runtime.h>

typedef __attribute__((ext_vector_type(2))) float v2f;
typedef __attribute__((ext_vector_type(8))) float v8f;

#define BATCH 2
#define NS 16384
#define NQ 8192
#define KNN 16
#define TILE 256
#define NWAVES 8   // 256 threads per block, wave32

// Branch-free bubble-carry insert into ascending sorted list.
// Per slot: 1 v_cmp + 4 v_cndmask — no nested ternaries, no short-circuit bools,
// so clang lowers to flat selects instead of divergent exec-mask branches.
__device__ __forceinline__ void insert16(float (&d)[KNN], int (&id)[KNN],
                                         float sc, int ni) {
  float cd = sc;
  int   ci = ni;
#pragma unroll
  for (int i = 0; i < KNN; ++i) {
    const float di = d[i];
    const int   ii = id[i];
    const bool  lt = cd < di;
    d[i]  = lt ? cd : di;
    id[i] = lt ? ci : ii;
    cd    = lt ? di : cd;
    ci    = lt ? ii : ci;
  }
}

// One block: 128 queries (8 waves x 16 queries each), scans all NS sources.
// WMMA mapping:  A (16x4, f32)  = source tile rows  (sx, sy, sz, s.s)
//                B (4x16, f32)  = query columns     (-2qx, -2qy, -2qz, 1)
//                D[m][n] = s.s - 2 q.s   (== d2 minus per-query constant q.q)
// C/D layout (wave32): lane L<16 : VGPR r -> M=r,   N=L
//                      lane L>=16: VGPR r -> M=r+8, N=L-16
// => every lane's 8 outputs belong to one query (L&15); top-k stays lane-private.
__global__ __launch_bounds__(256) void knn_wmma_kernel(
    const float* __restrict__ xyz,     // (B, NS, 3)
    const float* __restrict__ xyzq,    // (B, NQ, 3)
    int*  __restrict__ outIdx,         // (B, NQ, KNN)
    float* __restrict__ outDist)       // (B, NQ, KNN)
{
  __shared__ float4 tile[TILE];                 // staged source records (x,y,z,ss)
  __shared__ float  ldsD[NWAVES * 32 * KNN];    // per-lane sorted score lists
  __shared__ int    ldsI[NWAVES * 32 * KNN];

  const int tid   = threadIdx.x;
  const int lane  = tid & 31;
  const int wave  = tid >> 5;
  const bool hi   = lane >= 16;
  const int mrow  = lane & 15;                  // A-matrix source row this lane feeds

  const int qBlocksPerBatch = NQ / (NWAVES * 16);         // 64
  const int b      = blockIdx.x / qBlocksPerBatch;
  const int qBlock = blockIdx.x % qBlocksPerBatch;
  const int q      = qBlock * (NWAVES * 16) + wave * 16 + mrow;  // this lane's query

  // Load query point (both lane L and L+16 hold query L's data).
  const float* qp = xyzq + ((size_t)b * NQ + q) * 3;
  const float qx = qp[0], qy = qp[1], qz = qp[2];

  // B operand: VGPR0 = K0 row (lanes 0-15) / K1 row (lanes 16-31)
  //            VGPR1 = K2 row (lanes 0-15) / K3 row (lanes 16-31)
  v2f bvec;
  bvec.x = hi ? (-2.0f * qy) : (-2.0f * qx);
  bvec.y = hi ? 1.0f         : (-2.0f * qz);

  // Lane-private sorted (ascending) top-KNN of scores s.s - 2 q.s
  float d[KNN];
  int   id[KNN];
#pragma unroll
  for (int k = 0; k < KNN; ++k) { d[k] = 3.402823466e38f; id[k] = 0; }

  const float* sb = xyz + (size_t)b * NS * 3;

  for (int t0 = 0; t0 < NS; t0 += TILE) {
    // Cooperative stage: each thread builds one (x,y,z,ss) record in LDS.
    {
      const float* sp = sb + (size_t)(t0 + tid) * 3;
      const float sx = sp[0], sy = sp[1], sz = sp[2];
      const float ss = sx * sx + sy * sy + sz * sz;
      tile[tid] = make_float4(sx, sy, sz, ss);
      // Prefetch next tile's record into cache (global_prefetch_b8).
      if (t0 + TILE < NS)
        __builtin_prefetch(sb + (size_t)(t0 + TILE + tid) * 3, 0, 1);
    }
    __syncthreads();

    // Software-pipelined A-operand fetch: load step g+16 before consuming
    // the WMMA result of step g, hiding ds_load -> wmma latency behind the
    // threshold checks / inserts.
    const int half = hi ? 1 : 0;
    float2 av = ((const float2*)(&tile[mrow]))[half];

#pragma unroll 4
    for (int g = 0; g < TILE; g += 16) {
      v2f a; a.x = av.x; a.y = av.y;

      v8f c = {};
      c = __builtin_amdgcn_wmma_f32_16x16x4_f32(
              /*neg_a=*/false, a, /*neg_b=*/false, bvec,
              /*c_mod=*/(short)0, c, /*reuse_a=*/false, /*reuse_b=*/false);

      if (g + 16 < TILE)
        av = ((const float2*)(&tile[g + 16 + mrow]))[half];

      const int baseIdx = t0 + g + (hi ? 8 : 0);
#pragma unroll
      for (int r = 0; r < 8; ++r) {
        const float sc = c[r];
        if (sc < d[KNN - 1])            // threshold guard: insertion is rare
          insert16(d, id, sc, baseIdx + r);
      }
    }
    __syncthreads();
  }

  // Dump per-lane lists; lanes L and L+16 hold disjoint halves of query L's stream.
#pragma unroll
  for (int k = 0; k < KNN; ++k) {
    ldsD[(wave * 32 + lane) * KNN + k] = d[k];
    ldsI[(wave * 32 + lane) * KNN + k] = id[k];
  }
  __syncthreads();

  if (lane < 16) {
    const int baseA = (wave * 32 + lane) * KNN;
    const int baseB = (wave * 32 + lane + 16) * KNN;
    const float qq = qx * qx + qy * qy + qz * qz;
    const size_t outBase = (((size_t)b * NQ) + q) * KNN;
    int ia = 0, ib = 0;
    for (int k = 0; k < KNN; ++k) {
      const float da = ldsD[baseA + ia];
      const float db = ldsD[baseB + ib];
      const bool ta  = da <= db;
      const float ds = ta ? da : db;
      const int   is = ta ? ldsI[baseA + ia] : ldsI[baseB + ib];
      ia += ta ? 1 : 0;
      ib += ta ? 0 : 1;
      outIdx[outBase + k]  = is;
      outDist[outBase + k] = sqrtf(fmaxf(ds + qq, 0.0f));   // d2 = score + q.q
    }
  }
}

extern "C" void kernel_launch(void* const* d_in, const int* in_sizes, int n_in,
                              void* d_out, int out_size, void* d_ws, size_t ws_size,
                              hipStream_t stream) {
  const float* xyz  = (const float*)d_in[0];   // (B, NS, 3) f32
  const float* xyzq = (const float*)d_in[1];   // (B, NQ, 3) f32
  // d_out = [neighbors int32 (B,NQ,K)] ++ [distances f32 (B,NQ,K)]
  int*   outIdx  = (int*)d_out;
  float* outDist = (float*)d_out + (size_t)BATCH * NQ * KNN;

  dim3 grid(BATCH * (NQ / (NWAVES * 16)));     // 2 * 64 = 128 blocks
  dim3 block(256);
  knn_wmma_kernel<<<grid, block, 0, stream>>>(xyz, xyzq, outIdx, outDist);
}